// GraphSAGE_72129680769620
// MI455X (gfx1250) — compile-verified
//
#include <hip/hip_runtime.h>

typedef __attribute__((ext_vector_type(2))) float v2f;
typedef __attribute__((ext_vector_type(8))) float v8f;

#define WMMA_F32(A, B, C) \
    __builtin_amdgcn_wmma_f32_16x16x4_f32(false, (A), false, (B), (short)0, (C), false, false)

// ---------------------------------------------------------------------------
// Zero-fill scratch (grid-stride).
// ---------------------------------------------------------------------------
__global__ __launch_bounds__(256) void zero_f32(float* __restrict__ p, size_t n) {
    size_t i = (size_t)blockIdx.x * blockDim.x + threadIdx.x;
    size_t stride = (size_t)gridDim.x * blockDim.x;
    for (; i < n; i += stride) p[i] = 0.0f;
}

// ---------------------------------------------------------------------------
// In-degree counts: cnt[dst[e]] += 1.0f
// ---------------------------------------------------------------------------
__global__ __launch_bounds__(256) void edge_count(const int* __restrict__ dst,
                                                  float* __restrict__ cnt, int E) {
    int e = blockIdx.x * blockDim.x + threadIdx.x;
    if (e < E) atomicAdd(&cnt[dst[e]], 1.0f);
}

// ---------------------------------------------------------------------------
// Edge scatter-add: agg[dst[e], :] += feat[src[e], :]
// One wave32 per edge; contiguous float4 row gather (512B/wave/iter),
// global_atomic_add_f32 scatter (absorbed by the 192MB L2 — agg is 25-51MB).
// ---------------------------------------------------------------------------
__global__ __launch_bounds__(256) void edge_scatter(const int* __restrict__ src,
                                                    const int* __restrict__ dst,
                                                    const float* __restrict__ feat,
                                                    float* __restrict__ agg,
                                                    int E, int D) {
    int wavesPerBlock = blockDim.x >> 5;
    int gw = blockIdx.x * wavesPerBlock + (threadIdx.x >> 5);
    if (gw >= E) return;
    int lane = threadIdx.x & 31;
    int u = src[gw];
    int v = dst[gw];
    const float4* frow = (const float4*)(feat + (size_t)u * D);
    float* arow = agg + (size_t)v * D;
    int chunks = D >> 2;
    for (int c = lane; c < chunks; c += 32) {
        float4 val = frow[c];
        float* p = arow + (c << 2);
        atomicAdd(p + 0, val.x);
        atomicAdd(p + 1, val.y);
        atomicAdd(p + 2, val.z);
        atomicAdd(p + 3, val.w);
    }
}

// ---------------------------------------------------------------------------
// One K-step fragment set: A-self, A-neigh, 4x B-self, 4x B-neigh.
// ---------------------------------------------------------------------------
__device__ __forceinline__ void load_step(const float* __restrict__ xr,
                                          const float* __restrict__ ar,
                                          const float* __restrict__ wsB,
                                          const float* __restrict__ wnB,
                                          int tileStride, int k,
                                          v2f& a, v2f& an, v2f* bs, v2f* bn) {
    a  = *(const v2f*)(xr + k);
    an = *(const v2f*)(ar + k);
#pragma unroll
    for (int t = 0; t < 4; ++t) {
        bs[t] = *(const v2f*)(wsB + t * tileStride + k);
        bn[t] = *(const v2f*)(wnB + t * tileStride + k);
    }
}

__device__ __forceinline__ void compute_step(float invc, v2f a, v2f an,
                                             const v2f* bs, const v2f* bn,
                                             v8f* acc) {
    an.x *= invc;  // fused mean normalization (lowers to v_pk_mul_f32)
    an.y *= invc;
#pragma unroll
    for (int t = 0; t < 4; ++t) acc[t] = WMMA_F32(a, bs[t], acc[t]);
#pragma unroll
    for (int t = 0; t < 4; ++t) acc[t] = WMMA_F32(an, bn[t], acc[t]);
}

// ---------------------------------------------------------------------------
// Fused SAGE layer GEMM:
//   out[m,n] = ReLU( sum_k X[m,k]*Wself[n,k]
//                  + (1/max(cnt[m],1)) * sum_k AGG[m,k]*Wneigh[n,k]
//                  + bias[n] )
// fp32 WMMA (V_WMMA_F32_16X16X4_F32), wave32. Each wave: 16(M) x 64(N) tile,
// 4 accumulators, A fragment reused across the 4 N-tiles. blockDim = 64
// (2 waves => 16 x 128 per block).
// Requires O % 64 == 0 and D % 8 == 0 (true here: D in {128,256}, O in {256,128}).
//
// K loop unrolled x2 with ping-pong fragment sets: no rotation copies, and
// each WMMA group's operands were issued a half-iteration earlier, so the
// matrix pipe never waits on same-step loads.
//
// f32 16x16x4 fragment layout (ISA 7.12.2):
//   A (16x4):  lane L -> M = L&15, VGPR0/1 = K = 2*(L>>4) + {0,1}
//   B (4x16):  lane L -> N = L&15, VGPR0/1 = K = 2*(L>>4) + {0,1}
//   C/D:       VGPR r -> M = 8*(L>>4) + r, N = L&15
// ---------------------------------------------------------------------------
__global__ __launch_bounds__(64)
void sage_gemm_relu(const float* __restrict__ X,
                    const float* __restrict__ AGG,
                    const float* __restrict__ cnt,
                    const float* __restrict__ Wself,
                    const float* __restrict__ Wneigh,
                    const float* __restrict__ bias,
                    float* __restrict__ out,
                    int N, int D, int O) {
    const int lane = threadIdx.x & 31;
    const int wave = threadIdx.x >> 5;
    const int m0   = blockIdx.x * 16;
    const int col0 = (blockIdx.y * 2 + wave) * 64;
    if (col0 >= O) return;  // wave-uniform; EXEC stays all-1s inside

    const int nl = lane & 15;
    const int mA = min(m0 + nl, N - 1);       // A-fragment row for this lane
    const int kh = (lane >> 4) << 1;          // K sub-offset: 0 or 2

    const float invc = 1.0f / fmaxf(cnt[mA], 1.0f);

    const int n0 = col0 + nl;                 // B-fragment column for this lane
    const float* xr  = X   + (size_t)mA * D + kh;
    const float* ar  = AGG + (size_t)mA * D + kh;
    const float* wsB = Wself  + (size_t)n0 * D + kh;
    const float* wnB = Wneigh + (size_t)n0 * D + kh;
    const int tileStride = 16 * D;            // elements between N-tiles

    v8f acc[4] = {};

    // ping-pong fragment sets
    v2f a0, an0, bs0[4], bn0[4];
    v2f a1, an1, bs1[4], bn1[4];

    load_step(xr, ar, wsB, wnB, tileStride, 0, a0, an0, bs0, bn0);

    // main loop: two K-steps per iteration, D % 8 == 0
    for (int k = 0; k < D - 8; k += 8) {
        load_step(xr, ar, wsB, wnB, tileStride, k + 4, a1, an1, bs1, bn1);
        compute_step(invc, a0, an0, bs0, bn0, acc);
        load_step(xr, ar, wsB, wnB, tileStride, k + 8, a0, an0, bs0, bn0);
        compute_step(invc, a1, an1, bs1, bn1, acc);
    }
    // tail: steps D-8 (already in set 0) and D-4
    load_step(xr, ar, wsB, wnB, tileStride, D - 4, a1, an1, bs1, bn1);
    compute_step(invc, a0, an0, bs0, bn0, acc);
    compute_step(invc, a1, an1, bs1, bn1, acc);

    // Epilogue: bias + ReLU + store. C/D layout: VGPR r -> M = mb + r, N = nl.
    const int mb = m0 + ((lane >> 4) << 3);
    float bb[4];
#pragma unroll
    for (int t = 0; t < 4; ++t) bb[t] = bias[col0 + 16 * t + nl];
    for (int r = 0; r < 8; ++r) {
        int m = mb + r;
        if (m >= N) break;
        size_t o = (size_t)m * O + col0 + nl;
#pragma unroll
        for (int t = 0; t < 4; ++t)
            out[o + 16 * t] = fmaxf(acc[t][r] + bb[t], 0.0f);
    }
}

// ---------------------------------------------------------------------------
// Launch: zero scratch -> degree count -> scatter(x)->agg1 -> GEMM1 -> h1
//         -> scatter(h1)->agg2 -> GEMM2 -> out
// ---------------------------------------------------------------------------
extern "C" void kernel_launch(void* const* d_in, const int* in_sizes, int n_in,
                              void* d_out, int out_size, void* d_ws, size_t ws_size,
                              hipStream_t stream) {
    const float* x   = (const float*)d_in[0];
    const int*   ei  = (const int*)d_in[1];
    const float* Ws1 = (const float*)d_in[2];
    const float* Wn1 = (const float*)d_in[3];
    const float* b1  = (const float*)d_in[4];
    const float* Ws2 = (const float*)d_in[5];
    const float* Wn2 = (const float*)d_in[6];
    const float* b2  = (const float*)d_in[7];

    const int H     = in_sizes[4];            // 256
    const int F_IN  = in_sizes[2] / H;        // 128
    const int F_OUT = in_sizes[7];            // 128
    const int N     = in_sizes[0] / F_IN;     // 50000
    const int E     = in_sizes[1] / 2;        // 800000
    const int* src  = ei;
    const int* dst  = ei + E;

    // Scratch layout (floats): cnt | agg1 | agg2 | h1
    float* cnt  = (float*)d_ws;
    float* agg1 = cnt  + N;
    float* agg2 = agg1 + (size_t)N * F_IN;
    float* h1   = agg2 + (size_t)N * H;
    (void)ws_size;

    // Zero the atomically-accumulated regions (cnt, agg1, agg2 are contiguous).
    size_t zeroElems = (size_t)N * (size_t)(1 + F_IN + H);
    zero_f32<<<2048, 256, 0, stream>>>(cnt, zeroElems);

    // Degree counts (shared by both layers: same graph).
    edge_count<<<(E + 255) / 256, 256, 0, stream>>>(dst, cnt, E);

    // Layer 1 aggregation + fused GEMM/mean/bias/ReLU.
    edge_scatter<<<(E + 7) / 8, 256, 0, stream>>>(src, dst, x, agg1, E, F_IN);
    dim3 g1((N + 15) / 16, (H + 127) / 128);
    sage_gemm_relu<<<g1, 64, 0, stream>>>(x, agg1, cnt, Ws1, Wn1, b1, h1, N, F_IN, H);

    // Layer 2 aggregation + fused GEMM/mean/bias/ReLU -> d_out.
    edge_scatter<<<(E + 7) / 8, 256, 0, stream>>>(src, dst, h1, agg2, E, H);
    dim3 g2((N + 15) / 16, (F_OUT + 127) / 128);
    sage_gemm_relu<<<g2, 64, 0, stream>>>(h1, agg2, cnt, Ws2, Wn2, b2, (float*)d_out,
                                          N, H, F_OUT);
}